// LinearDeltaMemory_9208409882822
// MI455X (gfx1250) — compile-verified
//
#include <hip/hip_runtime.h>

// LinearDeltaMemory on MI455X (gfx1250):
//   Exact chunked WY reformulation of the delta-rule recurrence so the work
//   becomes f32 WMMA GEMMs on a 256KB LDS-resident state matrix M.
//   out_t = gamma_t * (M_c x_t + sum_{i<t} (k_i . x_t) M_c u_i)
//   M_next = gamma_C * (M_c + (M_c U^T) K),  u_j = r_j (x_j + sum_{i<j} G_ij u_i)
//   Frobenius clamp handled by a closed scalar recurrence on ||M||_F^2.

namespace {

constexpr int D_  = 256;   // feature dim
constexpr int S_  = 4096;  // sequence length
constexpr int CK  = 16;    // chunk length
constexpr int NCH = S_ / CK;
constexpr int MP  = 260;   // M row pitch (floats): stride-4 banks, 16B aligned
constexpr int XP  = 260;   // X/U row pitch
constexpr int ZP  = 18;    // Z row pitch
constexpr int GP  = 17;    // G row pitch
constexpr int NT  = 512;   // 16 wave32s per workgroup

typedef __attribute__((ext_vector_type(2))) float v2f;
typedef __attribute__((ext_vector_type(8))) float v8f;

__device__ __forceinline__ v8f wmma4(v2f a, v2f b, v8f c) {
  // D(16x16,f32) = A(16x4,f32) * B(4x16,f32) + C
  return __builtin_amdgcn_wmma_f32_16x16x4_f32(false, a, false, b, (short)0, c,
                                               false, false);
}

template <int XM>
__device__ __forceinline__ float swzx(float v) {
  // wave32 butterfly xor via ds_swizzle (group-of-32, xor_mask=XM, and=0x1f)
  return __int_as_float(
      __builtin_amdgcn_ds_swizzle(__float_as_int(v), (XM << 10) | 0x1f));
}

__global__ __launch_bounds__(NT, 1)
void ldm_chunk_kernel(const float* __restrict__ x,
                      const float* __restrict__ Minit,
                      const float* __restrict__ eta_raw,
                      const float* __restrict__ alpha_raw,
                      float* __restrict__ out,
                      float* __restrict__ mfinal)
{
  __shared__ __align__(16) float Ms[D_ * MP];   // state M, f32, LDS-resident
  __shared__ __align__(16) float Xs[CK * XP];   // chunk inputs (later scaled -> K)
  __shared__ __align__(16) float Us[CK * XP];   // UT vectors; reused as out-stage
  __shared__ __align__(16) float Zs[D_ * ZP];   // Z = M U^T
  __shared__ float Gs[CK * GP];                 // G_ij = (x_i . x_j)/n_i
  __shared__ float qpart[16 * 16];
  __shared__ float invn_s[CK], r_s[CK], coef_s[CK];
  __shared__ float q_s[CK];
  __shared__ float gam_s[CK + 1];
  __shared__ float F_s;                         // ||M||_F^2 carried across chunks

  const int tid  = threadIdx.x;
  const int lane = tid & 31;
  const int w    = tid >> 5;        // wave id 0..15
  const int hi   = lane >> 4;       // lane half (K-group select)
  const int ln   = lane & 15;       // row/col within tile

  const float eta   = 0.2f / (1.0f + __expf(-eta_raw[0]));
  const float alpha = 0.5f + 0.5f / (1.0f + __expf(-alpha_raw[0]));

  // ---- init: M_init -> LDS (coalesced), Frobenius^2 reduction ----
  float facc = 0.0f;
  for (int k = 0; k < 32; ++k) {
    int f = (tid << 2) + (k << 11);
    float4 v = *(const float4*)(Minit + f);
    *(float4*)(&Ms[(f >> 8) * MP + (f & 255)]) = v;
    facc += v.x * v.x + v.y * v.y + v.z * v.z + v.w * v.w;
  }
  facc += swzx<16>(facc); facc += swzx<8>(facc); facc += swzx<4>(facc);
  facc += swzx<2>(facc);  facc += swzx<1>(facc);
  if (lane == 0) qpart[w] = facc;

  // prefetch chunk 0 into registers (double-buffer across chunk loop)
  const float* xb = x + (size_t)blockIdx.x * S_ * D_;
  float4 pre0 = *(const float4*)(xb + (tid << 2));
  float4 pre1 = *(const float4*)(xb + (tid << 2) + 2048);
  __syncthreads();
  if (tid == 0) {
    float f = 0.0f;
    for (int i = 0; i < 16; ++i) f += qpart[i];
    F_s = f;
  }
  __syncthreads();

  for (int c = 0; c < NCH; ++c) {
    const int s0 = c * CK;
    // P0: stage prefetched X into LDS
    {
      int f0 = tid << 2;
      *(float4*)(&Xs[(f0 >> 8) * XP + (f0 & 255)]) = pre0;
      int f1 = f0 + 2048;
      *(float4*)(&Xs[(f1 >> 8) * XP + (f1 & 255)]) = pre1;
    }
    __syncthreads();
    if (c + 1 < NCH) {                    // hide HBM latency behind compute
      const float* xn = xb + (size_t)(s0 + CK) * D_;
      pre0 = *(const float4*)(xn + (tid << 2));
      pre1 = *(const float4*)(xn + (tid << 2) + 2048);
    }

    // P1a: per-row norms (wave w <-> chunk row w)
    {
      float ss = 0.0f;
      #pragma unroll
      for (int k = 0; k < 8; ++k) {
        float t = Xs[w * XP + lane + 32 * k];
        ss += t * t;
      }
      ss += swzx<16>(ss); ss += swzx<8>(ss); ss += swzx<4>(ss);
      ss += swzx<2>(ss);  ss += swzx<1>(ss);
      if (lane == 0) {
        float nx2 = ss;
        float n   = fmaxf(sqrtf(nx2), 1e-6f);
        float inv = 1.0f / n;
        float cc  = 1.0f - inv;
        invn_s[w] = inv;
        r_s[w]    = eta * cc / alpha;
        coef_s[w] = 2.0f * alpha * eta * cc * inv
                  + eta * eta * cc * cc * nx2 * inv * inv;
      }
    }
    // P1b: G = X X^T via WMMA (wave 15, deterministic single-writer)
    if (w == 15) {
      v8f g = {0,0,0,0,0,0,0,0};
      #pragma unroll 4
      for (int i0 = 0; i0 < D_; i0 += 4) {
        int k0 = i0 + 2 * hi;
        v2f a; a.x = Xs[ln * XP + k0]; a.y = Xs[ln * XP + k0 + 1];
        g = wmma4(a, a, g);            // A and B carry identical lane data
      }
      #pragma unroll
      for (int v = 0; v < 8; ++v) Gs[(v + 8 * hi) * GP + ln] = g[v];
    }
    // P1c: Y = M X^T (each wave: 16x16 tile, K=256 -> 64 WMMAs)
    v8f y = {0,0,0,0,0,0,0,0};
    {
      const float* mrow = &Ms[(16 * w + ln) * MP];
      const float* xrow = &Xs[ln * XP];
      #pragma unroll 4
      for (int i0 = 0; i0 < D_; i0 += 4) {
        int k0 = i0 + 2 * hi;
        v2f a;  a.x  = mrow[k0]; a.y  = mrow[k0 + 1];
        v2f bb; bb.x = xrow[k0]; bb.y = xrow[k0 + 1];
        y = wmma4(a, bb, y);
      }
    }
    __syncthreads();
    // P2: scale G rows by 1/n_i
    if (tid < 256) {
      int i = tid >> 4, j = tid & 15;
      Gs[i * GP + j] *= invn_s[i];
    }
    __syncthreads();
    // P3: UT transform, one thread per column d (fully register-resident)
    if (tid < 256) {
      float u[CK];
      #pragma unroll
      for (int j = 0; j < CK; ++j) {
        float acc = Xs[j * XP + tid];
        #pragma unroll
        for (int i = 0; i < j; ++i) acc += Gs[i * GP + j] * u[i];
        u[j] = r_s[j] * acc;
      }
      #pragma unroll
      for (int j = 0; j < CK; ++j) Us[j * XP + tid] = u[j];
    }
    __syncthreads();
    // P4a: Z = M U^T
    v8f z = {0,0,0,0,0,0,0,0};
    {
      const float* mrow = &Ms[(16 * w + ln) * MP];
      const float* urow = &Us[ln * XP];
      #pragma unroll 4
      for (int i0 = 0; i0 < D_; i0 += 4) {
        int k0 = i0 + 2 * hi;
        v2f a;  a.x  = mrow[k0]; a.y  = mrow[k0 + 1];
        v2f bb; bb.x = urow[k0]; bb.y = urow[k0 + 1];
        z = wmma4(a, bb, z);
      }
    }
    // P4b: spill own Z tile (needed as WMMA A-operand in the M update)
    #pragma unroll
    for (int v = 0; v < 8; ++v) Zs[(16 * w + v + 8 * hi) * ZP + ln] = z[v];
    // P4c: intra-chunk correction y[:,j] += sum_{i<j} G_ij * Z[:,i] (wave-local)
    #pragma unroll
    for (int i = 0; i < CK; ++i) {
      float g = Gs[i * GP + ln];
      g = (i < ln) ? g : 0.0f;
      #pragma unroll
      for (int v = 0; v < 8; ++v)
        y[v] += g * Zs[(16 * w + v + 8 * hi) * ZP + i];
    }
    // P4d: column-norm partials (deterministic, no float atomics)
    {
      float ssq = 0.0f;
      #pragma unroll
      for (int v = 0; v < 8; ++v) ssq += y[v] * y[v];
      ssq += swzx<16>(ssq);
      if (hi == 0) qpart[w * 16 + ln] = ssq;
    }
    __syncthreads();
    if (tid < 16) {
      float q = 0.0f;
      #pragma unroll
      for (int ww = 0; ww < 16; ++ww) q += qpart[ww * 16 + tid];
      q_s[tid] = q;
    }
    __syncthreads();
    // P5: scalar clamp/decay recurrence (gamma_j, F)
    if (tid == 0) {
      float F = F_s, gam = 1.0f;
      #pragma unroll
      for (int j = 0; j < CK; ++j) {
        gam_s[j] = gam;
        float P  = alpha * alpha * F + coef_s[j] * gam * gam * q_s[j];
        float sp = fminf(15.0f / (sqrtf(P) + 1e-6f), 1.0f);
        F   = sp * sp * P;
        gam *= sp * alpha;
      }
      gam_s[CK] = gam;
      F_s = F;
    }
    __syncthreads();
    // P6a: stage out_j = gamma_j * ytilde into Us[t][o] (transpose for coalescing)
    {
      float gc = gam_s[ln];
      #pragma unroll
      for (int v = 0; v < 8; ++v)
        Us[ln * XP + 16 * w + v + 8 * hi] = gc * y[v];
    }
    // P6b: X -> K in place (row scale by 1/n)
    {
      int row = tid >> 5;
      float inv = invn_s[row];
      int c0 = (tid & 31) << 3;
      #pragma unroll
      for (int k = 0; k < 8; ++k) Xs[row * XP + c0 + k] *= inv;
    }
    __syncthreads();
    // P7a: coalesced float4 output store
    {
      float* ob = out + ((size_t)blockIdx.x * S_ + s0) * D_;
      int f0 = tid << 2;
      *(float4*)(ob + f0) = *(const float4*)(&Us[(f0 >> 8) * XP + (f0 & 255)]);
      int f1 = f0 + 2048;
      *(float4*)(ob + f1) = *(const float4*)(&Us[(f1 >> 8) * XP + (f1 & 255)]);
    }
    // P7b: M = gammaC * (M + Z K), K=16 -> 4 WMMAs per 16x16 tile, 16 tiles/wave
    {
      const float gC = gam_s[CK];
      for (int ci = 0; ci < 16; ++ci) {
        int col = 16 * ci + ln;
        v8f acc;
        #pragma unroll
        for (int v = 0; v < 8; ++v)
          acc[v] = Ms[(16 * w + v + 8 * hi) * MP + col];
        #pragma unroll
        for (int kk = 0; kk < 4; ++kk) {
          int t0 = 4 * kk + 2 * hi;
          v2f a;  a.x  = Zs[(16 * w + ln) * ZP + t0];
                  a.y  = Zs[(16 * w + ln) * ZP + t0 + 1];
          v2f bb; bb.x = Xs[t0 * XP + col];
                  bb.y = Xs[(t0 + 1) * XP + col];
          acc = wmma4(a, bb, acc);
        }
        #pragma unroll
        for (int v = 0; v < 8; ++v)
          Ms[(16 * w + v + 8 * hi) * MP + col] = gC * acc[v];
      }
    }
    __syncthreads();
  }

  // ---- write M_final ----
  {
    float* mf = mfinal + (size_t)blockIdx.x * D_ * D_;
    for (int k = 0; k < 32; ++k) {
      int f = (tid << 2) + (k << 11);
      *(float4*)(mf + f) = *(const float4*)(&Ms[(f >> 8) * MP + (f & 255)]);
    }
  }
}

} // namespace

extern "C" void kernel_launch(void* const* d_in, const int* in_sizes, int n_in,
                              void* d_out, int out_size, void* d_ws, size_t ws_size,
                              hipStream_t stream) {
  const float* x         = (const float*)d_in[0];
  const float* Minit     = (const float*)d_in[1];
  const float* eta_raw   = (const float*)d_in[2];
  const float* alpha_raw = (const float*)d_in[3];
  const int B = in_sizes[0] / (S_ * D_);   // 16
  float* outp   = (float*)d_out;
  float* mfinal = outp + (size_t)B * S_ * D_;
  ldm_chunk_kernel<<<B, NT, 0, stream>>>(x, Minit, eta_raw, alpha_raw, outp, mfinal);
}